// GNNConvolution_70609262346585
// MI455X (gfx1250) — compile-verified
//
#include <hip/hip_runtime.h>
#include <hip/hip_bf16.h>
#include <math.h>

typedef float v2f __attribute__((ext_vector_type(2)));
typedef float v8f __attribute__((ext_vector_type(8)));

// D = A(16x4,f32) * B(4x16,f32) + C(16x16,f32), full fp32 precision on the matrix pipe.
#define WMMA_F32(a, b, c) \
    __builtin_amdgcn_wmma_f32_16x16x4_f32(false, (a), false, (b), (short)0, (c), false, false)

constexpr int kN = 50000;   // nodes
constexpr int kE = 800000;  // edges
constexpr int kD = 64;      // feature dim

// LDS weight layout: K-pair interleaved so a B fragment (W[krow][col], W[krow+1][col])
// is one contiguous 8-byte float2 -> single ds_load_b64 into an even VGPR pair.
//   sW[pair*128 + col*2 + (krow & 1)]  with pair = krow >> 1
__device__ __forceinline__ int wIdx(int row, int col) {
    return (row >> 1) * 128 + col * 2 + (row & 1);
}

// ---------------------------------------------------------------------------
// K1: fused node GEMM: q,k,v = x@W{q,k,v}+b ; agg = x@Ws+bs (skip branch);
//     also zero-inits per-node softmax max (encoded uint) and denom.
// One wave per 16-row tile; 4 weight matrices staged (interleaved) in LDS.
// ---------------------------------------------------------------------------
__global__ __launch_bounds__(256) void node_gemm_kernel(
    const float* __restrict__ x,
    const float* __restrict__ Wq, const float* __restrict__ bq,
    const float* __restrict__ Wk, const float* __restrict__ bk,
    const float* __restrict__ Wv, const float* __restrict__ bv,
    const float* __restrict__ Ws, const float* __restrict__ bs,
    float* __restrict__ q, float* __restrict__ k, float* __restrict__ v,
    float* __restrict__ agg, unsigned* __restrict__ nmax,
    float* __restrict__ denom, int nTiles)
{
    __shared__ float sW[4][kD * kD];
    __shared__ float sB[4][kD];
    const int t = threadIdx.x;
    for (int i = t; i < kD * kD; i += 256) {
        const int r = i >> 6, c = i & 63, d = wIdx(r, c);
        sW[0][d] = Wq[i]; sW[1][d] = Wk[i]; sW[2][d] = Wv[i]; sW[3][d] = Ws[i];
    }
    if (t < kD) { sB[0][t] = bq[t]; sB[1][t] = bk[t]; sB[2][t] = bv[t]; sB[3][t] = bs[t]; }
    __syncthreads();

    const int wave = t >> 5, lane = t & 31;
    const int M = lane & 15, kh = lane >> 4;   // wave32 WMMA fragment coords
    float* const outp[4] = {q, k, v, agg};

    for (int tile = blockIdx.x * 8 + wave; tile < nTiles; tile += gridDim.x * 8) {
        const int row0 = tile * 16;
        if (kh == 0) { nmax[row0 + M] = 0u; denom[row0 + M] = 0.0f; }

        v8f acc[4][4] = {};
        const float* xr = x + (size_t)(row0 + M) * kD;
        #pragma unroll 4
        for (int kk = 0; kk < 16; ++kk) {
            const int pair = 2 * kk + kh;        // K-pair index for this lane half
            const v2f a = *(const v2f*)(xr + 2 * pair);   // A 16x4 fragment
            #pragma unroll
            for (int m = 0; m < 4; ++m) {
                #pragma unroll
                for (int c = 0; c < 4; ++c) {
                    const v2f b = *(const v2f*)&sW[m][pair * 128 + (c * 16 + M) * 2];
                    acc[m][c] = WMMA_F32(a, b, acc[m][c]);
                }
            }
        }
        #pragma unroll
        for (int m = 0; m < 4; ++m) {
            #pragma unroll
            for (int c = 0; c < 4; ++c) {
                const float bias = sB[m][c * 16 + M];
                #pragma unroll
                for (int j = 0; j < 8; ++j) {    // C/D: VGPR j -> row j (+8 for hi lanes)
                    outp[m][(size_t)(row0 + j + 8 * kh) * kD + c * 16 + M] =
                        acc[m][c][j] + bias;
                }
            }
        }
    }
}

// ---------------------------------------------------------------------------
// K2: edge GEMM: e = edge_attr @ We   (no bias, matches PyG lin_edge)
// ---------------------------------------------------------------------------
__global__ __launch_bounds__(256) void edge_gemm_kernel(
    const float* __restrict__ ea, const float* __restrict__ We,
    float* __restrict__ e, int nTiles)
{
    __shared__ float sW[kD * kD];
    for (int i = threadIdx.x; i < kD * kD; i += 256)
        sW[wIdx(i >> 6, i & 63)] = We[i];
    __syncthreads();

    const int wave = threadIdx.x >> 5, lane = threadIdx.x & 31;
    const int M = lane & 15, kh = lane >> 4;

    for (int tile = blockIdx.x * 8 + wave; tile < nTiles; tile += gridDim.x * 8) {
        const int row0 = tile * 16;
        v8f acc[4] = {};
        const float* xr = ea + (size_t)(row0 + M) * kD;
        #pragma unroll 4
        for (int kk = 0; kk < 16; ++kk) {
            const int pair = 2 * kk + kh;
            const v2f a = *(const v2f*)(xr + 2 * pair);
            #pragma unroll
            for (int c = 0; c < 4; ++c) {
                const v2f b = *(const v2f*)&sW[pair * 128 + (c * 16 + M) * 2];
                acc[c] = WMMA_F32(a, b, acc[c]);
            }
        }
        #pragma unroll
        for (int c = 0; c < 4; ++c)
            #pragma unroll
            for (int j = 0; j < 8; ++j)
                e[(size_t)(row0 + j + 8 * kh) * kD + c * 16 + M] = acc[c][j];
    }
}

// ---------------------------------------------------------------------------
// K3: alpha = dot(q[dst], k[src]+e)/8 ; segment-max over dst via encoded atomicMax
// One wave (32 lanes) per edge; each lane handles a float2 of the 64 channels.
// ---------------------------------------------------------------------------
__global__ __launch_bounds__(256) void edge_alpha_kernel(
    const float* __restrict__ q, const float* __restrict__ k,
    const float* __restrict__ e, const int* __restrict__ src,
    const int* __restrict__ dst, float* __restrict__ alpha,
    unsigned* __restrict__ nmax, int nE)
{
    const int eid = blockIdx.x * 8 + (threadIdx.x >> 5);
    const int lane = threadIdx.x & 31;
    if (eid >= nE) return;
    const int s = src[eid], d = dst[eid];
    const float2 q2 = ((const float2*)(q + (size_t)d * kD))[lane];
    const float2 k2 = ((const float2*)(k + (size_t)s * kD))[lane];
    const float2 e2 = ((const float2*)(e + (size_t)eid * kD))[lane];
    float part = q2.x * (k2.x + e2.x) + q2.y * (k2.y + e2.y);
    #pragma unroll
    for (int off = 16; off > 0; off >>= 1) part += __shfl_xor(part, off, 32);
    if (lane == 0) {
        const float a = part * 0.125f;   // 1/sqrt(64)
        alpha[eid] = a;
        const unsigned bits = __float_as_uint(a);
        const unsigned key = (bits & 0x80000000u) ? ~bits : (bits | 0x80000000u);
        atomicMax(&nmax[d], key);
    }
}

// ---------------------------------------------------------------------------
// K4: p = exp(alpha - max[dst]) ; denom[dst] += p
// ---------------------------------------------------------------------------
__global__ __launch_bounds__(256) void edge_p_kernel(
    const float* __restrict__ alpha, const int* __restrict__ dst,
    const unsigned* __restrict__ nmax, float* __restrict__ p,
    float* __restrict__ denom, int nE)
{
    const int eid = blockIdx.x * blockDim.x + threadIdx.x;
    if (eid >= nE) return;
    const int d = dst[eid];
    const unsigned key = nmax[d];
    const unsigned bits = (key & 0x80000000u) ? (key & 0x7FFFFFFFu) : ~key;
    const float pv = expf(alpha[eid] - __uint_as_float(bits));
    p[eid] = pv;
    atomicAdd(&denom[d], pv);
}

// ---------------------------------------------------------------------------
// K5: agg[dst] += (v[src]+e) * p/denom[dst]   (wave per edge, float2 per lane)
// ---------------------------------------------------------------------------
__global__ __launch_bounds__(256) void edge_msg_kernel(
    const float* __restrict__ v, const float* __restrict__ e,
    const float* __restrict__ p, const float* __restrict__ denom,
    const int* __restrict__ src, const int* __restrict__ dst,
    float* __restrict__ agg, int nE)
{
    const int eid = blockIdx.x * 8 + (threadIdx.x >> 5);
    const int lane = threadIdx.x & 31;
    if (eid >= nE) return;
    const int s = src[eid], d = dst[eid];
    const float w = p[eid] / denom[d];
    const float2 v2 = ((const float2*)(v + (size_t)s * kD))[lane];
    const float2 e2 = ((const float2*)(e + (size_t)eid * kD))[lane];
    float* out = agg + (size_t)d * kD + lane * 2;
    atomicAdd(out,     (v2.x + e2.x) * w);
    atomicAdd(out + 1, (v2.y + e2.y) * w);
}

// ---------------------------------------------------------------------------
// K6: ReLU + LayerNorm (mode 0) or ReLU + log_softmax (mode 1); wave per node.
// ---------------------------------------------------------------------------
__global__ __launch_bounds__(256) void post_kernel(
    const float* __restrict__ agg, const float* __restrict__ gamma,
    const float* __restrict__ beta, float* __restrict__ out, int nN, int mode)
{
    const int node = blockIdx.x * 8 + (threadIdx.x >> 5);
    const int lane = threadIdx.x & 31;
    if (node >= nN) return;
    float2 x2 = ((const float2*)(agg + (size_t)node * kD))[lane];
    x2.x = fmaxf(x2.x, 0.0f);
    x2.y = fmaxf(x2.y, 0.0f);
    float2* op = (float2*)(out + (size_t)node * kD) + lane;
    if (mode == 0) {
        float s = x2.x + x2.y;
        #pragma unroll
        for (int off = 16; off > 0; off >>= 1) s += __shfl_xor(s, off, 32);
        const float mu = s * (1.0f / 64.0f);
        const float dx = x2.x - mu, dy = x2.y - mu;
        float vs = dx * dx + dy * dy;
        #pragma unroll
        for (int off = 16; off > 0; off >>= 1) vs += __shfl_xor(vs, off, 32);
        const float rs = rsqrtf(vs * (1.0f / 64.0f) + 1e-5f);
        float2 r;
        r.x = dx * rs * gamma[2 * lane]     + beta[2 * lane];
        r.y = dy * rs * gamma[2 * lane + 1] + beta[2 * lane + 1];
        *op = r;
    } else {
        float m = fmaxf(x2.x, x2.y);
        #pragma unroll
        for (int off = 16; off > 0; off >>= 1) m = fmaxf(m, __shfl_xor(m, off, 32));
        float s = expf(x2.x - m) + expf(x2.y - m);
        #pragma unroll
        for (int off = 16; off > 0; off >>= 1) s += __shfl_xor(s, off, 32);
        const float ls = m + logf(s);
        float2 r;
        r.x = x2.x - ls;
        r.y = x2.y - ls;
        *op = r;
    }
}

// ---------------------------------------------------------------------------
extern "C" void kernel_launch(void* const* d_in, const int* in_sizes, int n_in,
                              void* d_out, int out_size, void* d_ws, size_t ws_size,
                              hipStream_t stream) {
    (void)in_sizes; (void)n_in; (void)out_size; (void)ws_size;
    const float* node_attr = (const float*)d_in[0];
    const int*   eidx      = (const int*)d_in[1];
    const float* edge_attr = (const float*)d_in[2];
    const float* Wq = (const float*)d_in[3];
    const float* bq = (const float*)d_in[4];
    const float* Wk = (const float*)d_in[5];
    const float* bk = (const float*)d_in[6];
    const float* Wv = (const float*)d_in[7];
    const float* bv = (const float*)d_in[8];
    const float* We = (const float*)d_in[9];
    const float* Ws = (const float*)d_in[10];
    const float* bs = (const float*)d_in[11];
    const float* ln_g = (const float*)d_in[12];
    const float* ln_b = (const float*)d_in[13];

    const int* src = eidx;        // edge_index[0]
    const int* dst = eidx + kE;   // edge_index[1]

    // workspace carve (all f32, naturally aligned)
    float* q     = (float*)d_ws;
    float* k     = q    + (size_t)kN * kD;
    float* v     = k    + (size_t)kN * kD;
    float* agg   = v    + (size_t)kN * kD;
    float* xbuf  = agg  + (size_t)kN * kD;
    float* e     = xbuf + (size_t)kN * kD;
    float* alpha = e    + (size_t)kE * kD;
    float* p     = alpha + kE;
    float* denom = p + kE;
    unsigned* nmax = (unsigned*)(denom + kN);

    const int nodeTiles = kN / 16;   // 3125
    const int edgeTiles = kE / 16;   // 50000

    for (int layer = 0; layer < 3; ++layer) {
        const float* xin = (layer == 0) ? node_attr : xbuf;
        const size_t wo = (size_t)layer * kD * kD;
        const size_t bo = (size_t)layer * kD;

        node_gemm_kernel<<<200, 256, 0, stream>>>(
            xin, Wq + wo, bq + bo, Wk + wo, bk + bo, Wv + wo, bv + bo,
            Ws + wo, bs + bo, q, k, v, agg, nmax, denom, nodeTiles);

        edge_gemm_kernel<<<1000, 256, 0, stream>>>(edge_attr, We + wo, e, edgeTiles);

        edge_alpha_kernel<<<kE / 8, 256, 0, stream>>>(q, k, e, src, dst, alpha, nmax, kE);

        edge_p_kernel<<<(kE + 255) / 256, 256, 0, stream>>>(alpha, dst, nmax, p, denom, kE);

        edge_msg_kernel<<<kE / 8, 256, 0, stream>>>(v, e, p, denom, src, dst, agg, kE);

        if (layer < 2) {
            post_kernel<<<(kN + 7) / 8, 256, 0, stream>>>(
                agg, ln_g + bo, ln_b + bo, xbuf, kN, 0);
        } else {
            post_kernel<<<(kN + 7) / 8, 256, 0, stream>>>(
                agg, ln_g, ln_b, (float*)d_out, kN, 1);
        }
    }
}